// YOLOv2Loss_26027501814160
// MI455X (gfx1250) — compile-verified
//
#include <hip/hip_runtime.h>
#include <stdint.h>

#define NBATCH 16
#define SGRID  52
#define NANCH  5
#define NCLS   80
#define CELLVALS (NANCH * (5 + NCLS))        // 425 floats per cell per tensor
#define NCELLS   (NBATCH * SGRID * SGRID)    // 43264
#define WAVES_PER_BLOCK 8
#define NBLOCKS  (NCELLS / WAVES_PER_BLOCK)  // 5408 (divides exactly)

typedef __attribute__((ext_vector_type(2))) float v2f;
typedef __attribute__((ext_vector_type(8))) float v8f;

// ---------------- math helpers ----------------
__device__ __forceinline__ float sigmoidf_(float x) { return 1.0f / (1.0f + expf(-x)); }
__device__ __forceinline__ float softplusf_(float x) {
  return fmaxf(x, 0.0f) + log1pf(expf(-fabsf(x)));
}
__device__ __forceinline__ float anchW(int a) {
  float r = 0.57273f;
  r = (a == 1) ? 1.87446f : r;
  r = (a == 2) ? 3.33843f : r;
  r = (a == 3) ? 7.88282f : r;
  r = (a == 4) ? 9.77052f : r;
  return r;
}
__device__ __forceinline__ float anchH(int a) {
  float r = 0.677385f;
  r = (a == 1) ? 2.06253f : r;
  r = (a == 2) ? 5.47434f : r;
  r = (a == 3) ? 3.52778f : r;
  r = (a == 4) ? 9.16828f : r;
  return r;
}

// ---------------- CDNA5 async global->LDS streaming ----------------
__device__ __forceinline__ void async_b32_to_lds(uint32_t lds_off, const float* g) {
  asm volatile("global_load_async_to_lds_b32 %0, %1, off"
               :: "v"(lds_off), "v"((uint64_t)(uintptr_t)g)
               : "memory");
}
__device__ __forceinline__ void wait_async0() {
  asm volatile("s_wait_asynccnt 0" ::: "memory");
}

// ---------------- wave32 reductions ----------------
__device__ __forceinline__ float wave_max32(float v) {
  for (int o = 16; o > 0; o >>= 1) v = fmaxf(v, __shfl_xor(v, o, 32));
  return v;
}
__device__ __forceinline__ float wave_sum32(float v) {
  for (int o = 16; o > 0; o >>= 1) v += __shfl_xor(v, o, 32);
  return v;
}
__device__ __forceinline__ void wave_argmax32(float& v, int& idx) {
  for (int o = 16; o > 0; o >>= 1) {
    float vo = __shfl_xor(v, o, 32);
    int   io = __shfl_xor(idx, o, 32);
    if (vo > v || (vo == v && io < idx)) { v = vo; idx = io; }
  }
}

// Sum 32 per-lane f32 values with V_WMMA_F32_16X16X4_F32:
// A = ones(16x4), B = {v, 0} (4x16). Each D row = column sums of B, so the
// 32 lane values (one full B VGPR, bijective lane->entry striping) sum to
// sum_{n} D[0][n]; D row M=0 lives in c[0] lanes 0..15 (and M=8 copy in 16..31).
// EXEC must be all ones at the call site.
__device__ __forceinline__ float wave_sum_wmma(float v) {
  v2f a; a[0] = 1.0f; a[1] = 1.0f;
  v2f b; b[0] = v;    b[1] = 0.0f;
  v8f c = {};
  c = __builtin_amdgcn_wmma_f32_16x16x4_f32(false, a, false, b, (short)0, c, false, false);
  float s = c[0];
  for (int o = 8; o > 0; o >>= 1) s += __shfl_xor(s, o, 32); // reduce each 16-lane half
  return s;  // both halves hold the identical grand total
}

// ---------------- main kernel: one wave per cell ----------------
__global__ void __launch_bounds__(256) yolo_loss_main(
    const float* __restrict__ pred, const float* __restrict__ tgt,
    float* __restrict__ partial) {
  __shared__ __align__(16) float smem[WAVES_PER_BLOCK][2][432];
  __shared__ float red[WAVES_PER_BLOCK];

  const int lane = threadIdx.x & 31;
  const int wave = threadIdx.x >> 5;
  const int cell = blockIdx.x * WAVES_PER_BLOCK + wave;   // always < NCELLS
  const size_t base = (size_t)cell * CELLVALS;
  const float* gp = pred + base;
  const float* gt = tgt + base;
  float* sp = &smem[wave][0][0];
  float* st = &smem[wave][1][0];

  // Stream this cell's 425+425 floats into LDS via async loads (ASYNCcnt).
  for (int i = lane; i < CELLVALS; i += 32) {
    async_b32_to_lds((uint32_t)(uintptr_t)&sp[i], gp + i);
    async_b32_to_lds((uint32_t)(uintptr_t)&st[i], gt + i);
  }
  wait_async0();

  // ---- Phase A: 25 pairwise IoUs, one per lane (a = target anchor, p = pred anchor)
  float iou_l = -3.0f, tobj_l = 0.0f;
  if (lane < 25) {
    const int a = lane / 5, p = lane % 5;
    const float* tr = st + a * (5 + NCLS);
    const float* pr = sp + p * (5 + NCLS);
    const float gx = tr[0], gy = tr[1];
    const float gw = expf(tr[2]) * anchW(a), gh = expf(tr[3]) * anchH(a);
    tobj_l = tr[4];
    const float px = sigmoidf_(pr[0]), py = sigmoidf_(pr[1]);
    const float pw = expf(pr[2]) * anchW(p), ph = expf(pr[3]) * anchH(p);
    const float gx1 = gx - 0.5f * gw, gy1 = gy - 0.5f * gh;
    const float gx2 = gx + 0.5f * gw, gy2 = gy + 0.5f * gh;
    const float px1 = px - 0.5f * pw, py1 = py - 0.5f * ph;
    const float px2 = px + 0.5f * pw, py2 = py + 0.5f * ph;
    const float iw = fmaxf(fminf(gx2, px2) - fmaxf(gx1, px1), 0.0f);
    const float ih = fmaxf(fminf(gy2, py2) - fmaxf(gy1, py1), 0.0f);
    const float inter = iw * ih;
    const float uni = (gx2 - gx1) * (gy2 - gy1) + (px2 - px1) * (py2 - py1) - inter + 1e-9f;
    iou_l = inter / uni;
  }

  // ---- Greedy matching (redundant uniform computation on all lanes)
  unsigned mvec = 0xFFFFFu;   // 4-bit match slot per pred anchor, 0xF = unmatched
  unsigned takenMask = 0u;
  #pragma unroll
  for (int a = 0; a < NANCH; ++a) {
    const float vobj = __shfl(tobj_l, a * 5, 32);
    const bool valid = vobj > 0.5f;
    float bestv = -4.0f;
    int bestp = 0;
    #pragma unroll
    for (int p = 0; p < NANCH; ++p) {
      const float v = ((takenMask >> p) & 1u) ? -1.0f : __shfl(iou_l, a * 5 + p, 32);
      if (v > bestv) { bestv = v; bestp = p; }   // strict >: first-max like jnp.argmax
    }
    if (valid) {
      takenMask |= (1u << bestp);
      mvec = (mvec & ~(0xFu << (4 * bestp))) | ((unsigned)a << (4 * bestp));
    }
  }

  // ---- Phase B: per pred anchor slot, accumulate loss terms
  float acc = 0.0f;  // scalar terms accumulated on lane 0 only
  #pragma unroll
  for (int p = 0; p < NANCH; ++p) {
    const int m = (int)((mvec >> (4 * p)) & 0xFu);
    const float* pr = sp + p * (5 + NCLS);
    if (m < NANCH) {
      // matched -> obj: need class logsumexp + target-class argmax
      const float* tr = st + m * (5 + NCLS);
      const float v1 = pr[5 + lane];                              // classes 0..31
      const float v2 = pr[37 + lane];                             // classes 32..63
      const float v3 = (lane < 16) ? pr[69 + lane] : -1e30f;      // classes 64..79
      float mx = fmaxf(fmaxf(v1, v2), v3);
      mx = wave_max32(mx);
      float se = expf(v1 - mx) + expf(v2 - mx) + ((lane < 16) ? expf(v3 - mx) : 0.0f);
      se = wave_sum32(se);

      const float t1 = tr[5 + lane];
      const float t2 = tr[37 + lane];
      const float t3 = (lane < 16) ? tr[69 + lane] : -1e30f;
      float bv = t1; int bi = lane;
      if (t2 > bv) { bv = t2; bi = lane + 32; }
      if (t3 > bv) { bv = t3; bi = lane + 64; }
      wave_argmax32(bv, bi);                                      // all lanes get label

      const float nll = mx + logf(se) - pr[5 + bi];
      if (lane == 0) {
        const float sx = sigmoidf_(pr[0]), sy = sigmoidf_(pr[1]);
        const float tw = pr[2], th = pr[3], to = pr[4];
        const float dxy = (sx - tr[0]) * (sx - tr[0]) + (sy - tr[1]) * (sy - tr[1]);
        const float dwh = (tw - tr[2]) * (tw - tr[2]) + (th - tr[3]) * (th - tr[3]);
        acc += 5.0f * dxy + 5.0f * dwh + softplusf_(-to) + nll;
      }
    } else {
      if (lane == 0) acc += 0.5f * softplusf_(pr[4]);
    }
  }

  // ---- block reduction: 8 wave partials -> one per-block partial (WMMA path)
  if (lane == 0) red[wave] = acc;
  __syncthreads();
  if (wave == 0) {
    const float v = (lane < WAVES_PER_BLOCK) ? red[lane] : 0.0f;
    const float tot = wave_sum_wmma(v);   // full wave active here
    if (lane == 0) partial[blockIdx.x] = tot;
  }
}

// ---------------- final reduction kernel (single block, deterministic) ----------------
__global__ void __launch_bounds__(256) yolo_loss_reduce(
    const float* __restrict__ partial, int n, float* __restrict__ out) {
  __shared__ float red[8];
  const int tid = threadIdx.x;
  const int lane = tid & 31, wave = tid >> 5;
  float s = 0.0f;
  for (int i = tid; i < n; i += 256) s += partial[i];
  s = wave_sum32(s);
  if (lane == 0) red[wave] = s;
  __syncthreads();
  if (wave == 0) {
    const float v = (lane < 8) ? red[lane] : 0.0f;
    const float tot = wave_sum_wmma(v);
    if (lane == 0) out[0] = tot * (1.0f / (float)NBATCH);
  }
}

extern "C" void kernel_launch(void* const* d_in, const int* in_sizes, int n_in,
                              void* d_out, int out_size, void* d_ws, size_t ws_size,
                              hipStream_t stream) {
  const float* pred = (const float*)d_in[0];
  const float* tgt  = (const float*)d_in[1];
  float* out = (float*)d_out;
  float* partial = (float*)d_ws;   // needs NBLOCKS * 4 = 21632 bytes

  yolo_loss_main<<<NBLOCKS, 256, 0, stream>>>(pred, tgt, partial);
  yolo_loss_reduce<<<1, 256, 0, stream>>>(partial, NBLOCKS, out);
}